// MipmapWarp_369367187709
// MI455X (gfx1250) — compile-verified
//
#include <hip/hip_runtime.h>

#define NB 8
#define CH 16
#define HH 256
#define WW 256
#define MAXL 6

typedef __attribute__((ext_vector_type(2))) float v2f;
typedef __attribute__((ext_vector_type(8))) float v8f;

// ---------------------------------------------------------------------------
// Pyramid build: 2x blur-downsample, separable [1,3,3,1]/8 taps per axis.
// Horizontal pass in VALU -> LDS, vertical pass as banded matmul via
// V_WMMA_F32_16X16X4_F32 (9 K-chunks of 4, K padded to 36 with zeros).
// One wave (wave32) per 16x16 output tile.
// ---------------------------------------------------------------------------

__device__ __forceinline__ int reflect_clamp(int i, int n) {
  if (i < 0) i = -i;                 // pad=1 reflect: -1 -> 1
  if (i >= n) i = 2 * n - 2 - i;     // n -> n-2
  if (i < 0) i = 0;
  if (i > n - 1) i = n - 1;          // safety clamp for tile overrun
  return i;
}

__device__ __forceinline__ float gtap(int d) {
  float w = 0.0f;
  if (d == 0 || d == 3) w = 0.125f;
  if (d == 1 || d == 2) w = 0.375f;
  return w;
}

__global__ void __launch_bounds__(128) downsample2x_wmma(
    const float* __restrict__ src, float* __restrict__ dst,
    int h, int w, int tilesX, int tilesY) {
  __shared__ float P[4][34 * 36];  // input patch, padded row stride
  __shared__ float T[4][36 * 16];  // horizontal result, K-padded to 36 rows

  const int lane = threadIdx.x & 31;
  const int wv = threadIdx.x >> 5;
  const int tilesPerImg = tilesX * tilesY;
  const int gw = blockIdx.x * 4 + wv;     // global wave id == tile id
  const int img = gw / tilesPerImg;       // (n*C + c) image index, 0..127
  const int t = gw % tilesPerImg;
  const int ty0 = (t / tilesX) * 16;
  const int tx0 = (t % tilesX) * 16;
  const int h2 = h >> 1, w2 = w >> 1;
  const float* s = src + (size_t)img * h * w;
  float* d = dst + (size_t)img * h2 * w2;

  // Phase 1: load 34x34 reflect-padded patch into LDS
  for (int it = 0; it < 37; ++it) {
    int e = it * 32 + lane;
    if (e < 34 * 34) {
      int r = e / 34, c = e % 34;
      int iy = reflect_clamp(2 * ty0 - 1 + r, h);
      int ix = reflect_clamp(2 * tx0 - 1 + c, w);
      P[wv][r * 36 + c] = s[iy * w + ix];
    }
  }
  __syncthreads();

  // Phase 2: horizontal blur + decimate -> T[36][16] (rows 34,35 = 0)
  for (int it = 0; it < 18; ++it) {
    int e = it * 32 + lane;  // 576 entries exactly
    int r = e >> 4, j = e & 15;
    float v = 0.0f;
    if (r < 34) {
      const float* pr = &P[wv][r * 36 + 2 * j];
      v = 0.125f * pr[0] + 0.375f * pr[1] + 0.375f * pr[2] + 0.125f * pr[3];
    }
    T[wv][r * 16 + j] = v;
  }
  __syncthreads();

  // Phase 3: vertical banded matmul Out(16x16) = G(16x36) * T(36x16)
  // A layout (16x4 f32): lanes 0-15 -> K=0,1 ; lanes 16-31 -> K=2,3
  const int Mrow = lane & 15;
  const int kb = (lane >> 4) << 1;  // 0 or 2
  v8f acc = {};
#pragma unroll
  for (int kk = 0; kk < 9; ++kk) {
    int r0 = 4 * kk + kb;
    v2f a, b;
    a.x = gtap(r0 - 2 * Mrow);        // G[i][r] = g[r - 2i], banded
    a.y = gtap(r0 + 1 - 2 * Mrow);
    b.x = T[wv][r0 * 16 + Mrow];      // B[k][n], n = lane&15
    b.y = T[wv][(r0 + 1) * 16 + Mrow];
    acc = __builtin_amdgcn_wmma_f32_16x16x4_f32(
        false, a, false, b, (short)0, acc, false, false);
  }

  // Store: C/D layout — VGPR v: lanes0-15 M=v, lanes16-31 M=v+8
#pragma unroll
  for (int v = 0; v < 8; ++v) {
    int row = ty0 + ((lane >> 4) << 3) + v;
    int col = tx0 + Mrow;
    if (row < h2 && col < w2) d[row * w2 + col] = acc[v];
  }
}

// ---------------------------------------------------------------------------
// Warp / gather kernel: mip-level selection + double-bilinear mip sampling.
// Upsample folded analytically: U_l(yi,xi) = bilinear tap of low-res level at
// src = (i+0.5)/2^l - 0.5 (clipped), so no full-res stack is materialized.
// ---------------------------------------------------------------------------

__device__ __forceinline__ int lvl_off(int l) {
  switch (l) {
    case 1:  return 0;
    case 2:  return 2097152;
    case 3:  return 2621440;
    case 4:  return 2752512;
    default: return 2785280;
  }
}

__device__ __forceinline__ float bilerp(const float* __restrict__ p, int w,
                                        int y0, int y1, int x0, int x1,
                                        float fy, float fx) {
  float a = p[y0 * w + x0], b = p[y0 * w + x1];
  float c = p[y1 * w + x0], d = p[y1 * w + x1];
  float top = a + fx * (b - a);
  float bot = c + fx * (d - c);
  return top + fy * (bot - top);
}

struct LS {
  int off_n;  // ws offset for (level, n, channel 0)
  int sz;     // hl*wl
  int wl;
  int fy0[2], fy1[2], fx0[2], fx1[2];
  float by[2], ax[2];
  bool zero;
};

__device__ __forceinline__ void make_ls(LS& L, int l, int n, int y0i, int y1i,
                                        int x0i, int x1i) {
  if (l == 0) { L.zero = true; return; }
  L.zero = false;
  int hl = HH >> l, wl = WW >> l;
  L.wl = wl;
  L.sz = hl * wl;
  L.off_n = lvl_off(l) + n * CH * L.sz;
  float invs = 1.0f / (float)(1 << l);
  int ys[2] = {y0i, y1i};
  int xs[2] = {x0i, x1i};
#pragma unroll
  for (int j = 0; j < 2; ++j) {
    float sy = ((float)ys[j] + 0.5f) * invs - 0.5f;
    sy = fminf(fmaxf(sy, 0.0f), (float)(hl - 1));
    float f0 = floorf(sy);
    L.fy0[j] = (int)f0;
    L.fy1[j] = min(L.fy0[j] + 1, hl - 1);
    L.by[j] = sy - f0;
    float sx = ((float)xs[j] + 0.5f) * invs - 0.5f;
    sx = fminf(fmaxf(sx, 0.0f), (float)(wl - 1));
    float g0 = floorf(sx);
    L.fx0[j] = (int)g0;
    L.fx1[j] = min(L.fx0[j] + 1, wl - 1);
    L.ax[j] = sx - g0;
  }
}

__device__ __forceinline__ float sample_ls(const LS& L,
                                           const float* __restrict__ inp_nc,
                                           const float* __restrict__ ws, int c,
                                           int y0i, int y1i, int x0i, int x1i,
                                           float wy, float wx) {
  if (L.zero) return bilerp(inp_nc, WW, y0i, y1i, x0i, x1i, wy, wx);
  const float* p = ws + L.off_n + c * L.sz;
  float U00 = bilerp(p, L.wl, L.fy0[0], L.fy1[0], L.fx0[0], L.fx1[0], L.by[0], L.ax[0]);
  float U01 = bilerp(p, L.wl, L.fy0[0], L.fy1[0], L.fx0[1], L.fx1[1], L.by[0], L.ax[1]);
  float U10 = bilerp(p, L.wl, L.fy0[1], L.fy1[1], L.fx0[0], L.fx1[0], L.by[1], L.ax[0]);
  float U11 = bilerp(p, L.wl, L.fy0[1], L.fy1[1], L.fx0[1], L.fx1[1], L.by[1], L.ax[1]);
  float top = U00 + wx * (U01 - U00);
  float bot = U10 + wx * (U11 - U10);
  return top + wy * (bot - top);
}

__global__ void __launch_bounds__(256) mipwarp(
    const float* __restrict__ inp, const float* __restrict__ grid,
    const float* __restrict__ ws, float* __restrict__ out) {
  int pix = blockIdx.x * blockDim.x + threadIdx.x;
  if (pix >= NB * HH * WW) return;
  int n = pix / (HH * WW);
  int rem = pix - n * (HH * WW);
  int y = rem / WW;
  int x = rem - y * WW;
  const float* gn = grid + (size_t)n * HH * WW * 2;
  float gx = gn[(y * WW + x) * 2 + 0];
  float gy = gn[(y * WW + x) * 2 + 1];

  // Mip level from max neighbor distance (edge-clamped neighbors)
  float xc = (WW - 1) * (gx + 1.0f) * 0.5f;
  float yc = (HH - 1) * (gy + 1.0f) * 0.5f;
  float dmax2 = 0.0f;
  int nx[4] = {max(x - 1, 0), min(x + 1, WW - 1), x, x};
  int ny[4] = {y, y, max(y - 1, 0), min(y + 1, HH - 1)};
#pragma unroll
  for (int k = 0; k < 4; ++k) {
    float ggx = gn[(ny[k] * WW + nx[k]) * 2 + 0];
    float ggy = gn[(ny[k] * WW + nx[k]) * 2 + 1];
    float oxc = (WW - 1) * (ggx + 1.0f) * 0.5f;
    float oyc = (HH - 1) * (ggy + 1.0f) * 0.5f;
    float dx = oxc - xc, dy = oyc - yc;
    dmax2 = fmaxf(dmax2, dx * dx + dy * dy);
  }
  float dd = sqrtf(fmaxf(dmax2, 1.0f));
  float lev = fminf(fmaxf(__log2f(dd), 0.0f), (float)(MAXL - 1));
  int l0 = (int)floorf(lev);
  int l1 = (int)ceilf(lev);
  float lw = lev - floorf(lev);

  // Grid-sample coords (align_corners=False style, border clamp)
  float sxp = ((gx + 1.0f) * WW - 1.0f) * 0.5f;
  float syp = ((gy + 1.0f) * HH - 1.0f) * 0.5f;
  sxp = fminf(fmaxf(sxp, 0.0f), (float)(WW - 1));
  syp = fminf(fmaxf(syp, 0.0f), (float)(HH - 1));
  float fx0 = floorf(sxp), fy0 = floorf(syp);
  float wx = sxp - fx0, wy = syp - fy0;
  int x0i = (int)fx0, y0i = (int)fy0;
  int x1i = min(x0i + 1, WW - 1);
  int y1i = min(y0i + 1, HH - 1);

  LS L0, L1;
  make_ls(L0, l0, n, y0i, y1i, x0i, x1i);
  make_ls(L1, l1, n, y0i, y1i, x0i, x1i);
  bool same = (l1 == l0);

  for (int c = 0; c < CH; ++c) {
    const float* inc = inp + ((size_t)(n * CH + c)) * HH * WW;
    float v0 = sample_ls(L0, inc, ws, c, y0i, y1i, x0i, x1i, wy, wx);
    float v1 = same ? v0 : sample_ls(L1, inc, ws, c, y0i, y1i, x0i, x1i, wy, wx);
    out[((size_t)(n * CH + c) * HH + y) * WW + x] = v0 + lw * (v1 - v0);
  }
}

// ---------------------------------------------------------------------------

extern "C" void kernel_launch(void* const* d_in, const int* in_sizes, int n_in,
                              void* d_out, int out_size, void* d_ws,
                              size_t ws_size, hipStream_t stream) {
  const float* inp = (const float*)d_in[0];   // (8,16,256,256) f32
  const float* grid = (const float*)d_in[1];  // (8,256,256,2) f32
  float* out = (float*)d_out;                 // (8,16,256,256) f32
  float* ws = (float*)d_ws;                   // pyramid: ~11.2 MB of f32

  float* P1 = ws + 0;        // 128 x 128x128
  float* P2 = ws + 2097152;  // 128 x 64x64
  float* P3 = ws + 2621440;  // 128 x 32x32
  float* P4 = ws + 2752512;  // 128 x 16x16
  float* P5 = ws + 2785280;  // 128 x 8x8

  // blocks = (128 images * tilesPerImg waves) / 4 waves per block
  downsample2x_wmma<<<(128 * 64) / 4, 128, 0, stream>>>(inp, P1, 256, 256, 8, 8);
  downsample2x_wmma<<<(128 * 16) / 4, 128, 0, stream>>>(P1, P2, 128, 128, 4, 4);
  downsample2x_wmma<<<(128 * 4) / 4, 128, 0, stream>>>(P2, P3, 64, 64, 2, 2);
  downsample2x_wmma<<<(128 * 1) / 4, 128, 0, stream>>>(P3, P4, 32, 32, 1, 1);
  downsample2x_wmma<<<(128 * 1) / 4, 128, 0, stream>>>(P4, P5, 16, 16, 1, 1);

  mipwarp<<<(NB * HH * WW + 255) / 256, 256, 0, stream>>>(inp, grid, ws, out);
}